// FFF_v2_17222818857440
// MI455X (gfx1250) — compile-verified
//
#include <hip/hip_runtime.h>

// CDNA5 / gfx1250: wave32, WMMA f32 16x16x4 for the full-precision router GEMM.

typedef __attribute__((ext_vector_type(2))) float v2f;
typedef __attribute__((ext_vector_type(8))) float v8f;

#define NIN        4096
#define NOUT       4096
#define TREE_DEPTH 12
#define B_ROWS     8192
#define LAM_STRIDE 16   // lam stored padded [B][16], cols 12..15 are zero

// ---------------------------------------------------------------------------
// Kernel 1: lam[B,16] = x[B,4096] @ W.T (W is [12,4096]; N padded to 16)
// One wave computes one 16x16 tile of lam via V_WMMA_F32_16X16X4_F32.
// A fragment (16x4 f32): lane m = lane&15 holds row M=m; khi = lane>>4 selects
//   K-half: VGPR0 = A[m, k+2*khi], VGPR1 = A[m, k+2*khi+1].
// B fragment (4x16 f32): n = lane&15 is column N; VGPR0 = B[2*khi, n] = W[n][k+2*khi],
//   VGPR1 = W[n][k+2*khi+1]. Columns n>=12 are zero-masked (keep=0).
// C/D (16x16 f32): VGPR r -> row M = r + 8*khi, col N = lane&15.
// ---------------------------------------------------------------------------
__global__ __launch_bounds__(256)
void router_wmma_kernel(const float* __restrict__ x,
                        const float* __restrict__ W,
                        float* __restrict__ lam) {
    const int lane = threadIdx.x & 31;
    const int wave = threadIdx.x >> 5;
    const int tile = blockIdx.x * 8 + wave;      // 16 batch rows per tile
    const int m    = lane & 15;                  // A row (M) / B column (N)
    const int khi  = lane >> 4;                  // K-half selector

    const float* __restrict__ xrow = x + (size_t)(tile * 16 + m) * NIN + 2 * khi;
    const float keep = (m < TREE_DEPTH) ? 1.0f : 0.0f;
    const float* __restrict__ wrow =
        W + (size_t)((m < TREE_DEPTH) ? m : 0) * NIN + 2 * khi;

    v8f c = {0.f, 0.f, 0.f, 0.f, 0.f, 0.f, 0.f, 0.f};

    #pragma unroll 8
    for (int k = 0; k < NIN; k += 4) {
        v2f a, b;
        a.x = xrow[k];
        a.y = xrow[k + 1];
        b.x = keep * wrow[k];
        b.y = keep * wrow[k + 1];
        // 8 args: (neg_a, A, neg_b, B, c_mod, C, reuse_a, reuse_b)
        c = __builtin_amdgcn_wmma_f32_16x16x4_f32(
                false, a, false, b, (short)0, c, false, false);
    }

    float* __restrict__ dst =
        lam + (size_t)(tile * 16 + khi * 8) * LAM_STRIDE + m;
    #pragma unroll
    for (int r = 0; r < 8; ++r) {
        dst[(size_t)r * LAM_STRIDE] = c[r];
    }
}

// ---------------------------------------------------------------------------
// Kernel 2: out[b] = sum_i lam[b,i] * Y[node(b,i)]
// node_i = (2^i - 1) + prefix, prefix' = 2*prefix + (lam_i > 0).
// One block per batch row; 256 threads x 4 float4 chunks cover 4096 cols.
// Y (67 MB) is L2-resident on MI455X (192 MB L2) -> gathers hit L2.
// ---------------------------------------------------------------------------
__global__ __launch_bounds__(256)
void tree_gather_kernel(const float* __restrict__ lam,
                        const float* __restrict__ Y,
                        float* __restrict__ out) {
    const int b = blockIdx.x;
    const int t = threadIdx.x;

    float l[TREE_DEPTH];
    int   node[TREE_DEPTH];
    const float* __restrict__ lamb = lam + (size_t)b * LAM_STRIDE;
    int prefix = 0;
    #pragma unroll
    for (int i = 0; i < TREE_DEPTH; ++i) {
        l[i]    = lamb[i];
        node[i] = (1 << i) - 1 + prefix;
        prefix  = 2 * prefix + (l[i] > 0.0f ? 1 : 0);
    }

    // Prefetch the 12 gathered rows (cachelines this thread will touch).
    #pragma unroll
    for (int i = 0; i < TREE_DEPTH; ++i) {
        __builtin_prefetch(Y + (size_t)node[i] * NOUT + t * 4, 0, 3);
    }

    float4 acc[4];
    #pragma unroll
    for (int ch = 0; ch < 4; ++ch) {
        acc[ch].x = 0.f; acc[ch].y = 0.f; acc[ch].z = 0.f; acc[ch].w = 0.f;
    }

    #pragma unroll
    for (int i = 0; i < TREE_DEPTH; ++i) {
        const float coef = l[i];
        const float* __restrict__ yr = Y + (size_t)node[i] * NOUT;
        #pragma unroll
        for (int ch = 0; ch < 4; ++ch) {
            const float4 v =
                *reinterpret_cast<const float4*>(yr + ch * 1024 + t * 4);
            acc[ch].x += coef * v.x;
            acc[ch].y += coef * v.y;
            acc[ch].z += coef * v.z;
            acc[ch].w += coef * v.w;
        }
    }

    float* __restrict__ o = out + (size_t)b * NOUT;
    #pragma unroll
    for (int ch = 0; ch < 4; ++ch) {
        *reinterpret_cast<float4*>(o + ch * 1024 + t * 4) = acc[ch];
    }
}

extern "C" void kernel_launch(void* const* d_in, const int* in_sizes, int n_in,
                              void* d_out, int out_size, void* d_ws, size_t ws_size,
                              hipStream_t stream) {
    (void)in_sizes; (void)n_in; (void)out_size; (void)ws_size;

    const float* x = (const float*)d_in[0];   // [8192, 4096]
    const float* W = (const float*)d_in[1];   // [12, 4096]
    const float* Y = (const float*)d_in[2];   // [4095, 4096]
    float* out = (float*)d_out;               // [8192, 4096]
    float* lam = (float*)d_ws;                // [8192, 16] padded, 512 KB

    // 8192/16 = 512 tiles, 8 waves (one tile each) per 256-thread block.
    router_wmma_kernel<<<dim3(B_ROWS / 16 / 8), dim3(256), 0, stream>>>(x, W, lam);
    // One block per batch row.
    tree_gather_kernel<<<dim3(B_ROWS), dim3(256), 0, stream>>>(lam, Y, out);
}